// SSIMLoss_69295002353718
// MI455X (gfx1250) — compile-verified
//
#include <hip/hip_runtime.h>

typedef float v2f __attribute__((ext_vector_type(2)));
typedef float v8f __attribute__((ext_vector_type(8)));

#define H_IMG 512
#define W_IMG 512
#define TILE 16
#define HALO 5
#define INW 26                 // TILE + 2*HALO
#define NPLANES 48             // 16 * 3
#define WAVES_PER_BLOCK 4
#define NTILES (NPLANES * 32 * 32)            // 49152
#define NBLOCKS (NTILES / WAVES_PER_BLOCK)    // 12288

// banded separable-gaussian weight: W[j][c] = w1d[c-j] for c-j in [0,10], else 0
__device__ __forceinline__ float bandw(int j, int c, float invS) {
    int d = c - j;
    float x = (float)(d - 5);
    float v = __expf(x * x * (-1.0f / 4.5f)) * invS;   // sigma=1.5 -> 2*sigma^2 = 4.5
    return (d >= 0 && d <= 10) ? v : 0.0f;
}

__global__ __launch_bounds__(128) void ssim_wmma_kernel(
        const float* __restrict__ pred,
        const float* __restrict__ targ,
        double* __restrict__ acc) {
    __shared__ __align__(16) float sIn[WAVES_PER_BLOCK][2][INW * INW];
    __shared__ __align__(16) float sH[WAVES_PER_BLOCK][5][32 * 16];

    const int lane = threadIdx.x & 31;
    const int wv   = threadIdx.x >> 5;
    const int half = lane >> 4;   // which 16-lane half
    const int n    = lane & 15;   // column / M index within half

    const int tileIdx = blockIdx.x * WAVES_PER_BLOCK + wv;
    const int plane = tileIdx >> 10;          // 1024 tiles per 512x512 plane
    const int tp    = tileIdx & 1023;
    const int ty = tp >> 5, tx = tp & 31;
    const int y0 = ty * TILE, x0 = tx * TILE;

    const float* P = pred + (size_t)plane * (H_IMG * W_IMG);
    const float* T = targ + (size_t)plane * (H_IMG * W_IMG);

    // normalized 1D gaussian
    float S = 0.f;
#pragma unroll
    for (int i = 0; i < 11; ++i) {
        float x = (float)(i - 5);
        S += __expf(x * x * (-1.0f / 4.5f));
    }
    const float invS = 1.0f / S;

    // Banded weight chunks. Per-lane values are identical for the A-slot
    // (16x4: lane -> M = lane&15, K = 4t + 2*half (+1)) and the B-slot
    // (4x16: VGPR0 -> K rows 4t+2*half, VGPR1 -> +1; lane -> N = lane&15).
    v2f wband[7];
#pragma unroll
    for (int t = 0; t < 7; ++t) {
        int c0 = 4 * t + 2 * half;
        wband[t].x = bandw(n, c0, invS);
        wband[t].y = bandw(n, c0 + 1, invS);
    }

    // Stage zero-padded 26x26 neighborhoods of pred/target into LDS.
    for (int i = lane; i < INW * INW; i += 32) {
        int r = i / INW;
        int c = i - r * INW;
        int gy = y0 - HALO + r;
        int gx = x0 - HALO + c;
        float vp = 0.f, vt = 0.f;
        if ((unsigned)gy < (unsigned)H_IMG && (unsigned)gx < (unsigned)W_IMG) {
            int off = gy * W_IMG + gx;
            vp = P[off];
            vt = T[off];
        }
        sIn[wv][0][i] = vp;
        sIn[wv][1][i] = vt;
    }
    __syncthreads();

    // ---- horizontal pass: H_q(26x16) = In_q(26x26) x Wband(26x16), 2 row groups.
    // A-chunks for p and t are loaded once per row group and reused by all 5
    // quantities (p, t, p*p, t*t, p*t built with elementwise multiplies).
#pragma unroll
    for (int rg = 0; rg < 2; ++rg) {
        int m = n + rg * 16;
        bool rowOk = (m < INW);
        int mm = rowOk ? m : 0;
        v2f Ap[7], At[7];
#pragma unroll
        for (int t = 0; t < 7; ++t) {
            int c0 = 4 * t + 2 * half;
            bool ok = rowOk && (c0 + 1 < INW);  // c0 even; kills the 26/27 pad cols
            int cc = ok ? c0 : 0;
            const float* bp = &sIn[wv][0][mm * INW + cc];
            const float* bt = &sIn[wv][1][mm * INW + cc];
            float p0 = bp[0], p1 = bp[1];
            float t0 = bt[0], t1 = bt[1];
            if (!ok) { p0 = p1 = t0 = t1 = 0.f; }
            v2f ap = {p0, p1};
            v2f at = {t0, t1};
            Ap[t] = ap;
            At[t] = at;
        }
#pragma unroll
        for (int q = 0; q < 5; ++q) {
            v8f hacc = {0.f, 0.f, 0.f, 0.f, 0.f, 0.f, 0.f, 0.f};
#pragma unroll
            for (int t = 0; t < 7; ++t) {
                v2f a;
                if      (q == 0) a = Ap[t];
                else if (q == 1) a = At[t];
                else if (q == 2) a = Ap[t] * Ap[t];
                else if (q == 3) a = At[t] * At[t];
                else             a = Ap[t] * At[t];
                hacc = __builtin_amdgcn_wmma_f32_16x16x4_f32(
                    false, a, false, wband[t], (short)0, hacc, false, false);
            }
            // D layout: VGPR v -> rows (rg*16 + v) lanes0-15, (rg*16 + v + 8) lanes16-31
#pragma unroll
            for (int v = 0; v < 8; ++v) {
                int row = rg * 16 + v + 8 * half;
                sH[wv][q][row * 16 + n] = hacc[v];
            }
        }
    }
    __syncthreads();

    // ---- vertical pass: Out_q(16x16) = Wband(16x26) x H_q(26x16)
    v8f res[5];
#pragma unroll
    for (int q = 0; q < 5; ++q) {
        v8f vacc = {0.f, 0.f, 0.f, 0.f, 0.f, 0.f, 0.f, 0.f};
#pragma unroll
        for (int t = 0; t < 7; ++t) {
            int r0 = 4 * t + 2 * half;
            v2f b = { sH[wv][q][r0 * 16 + n], sH[wv][q][(r0 + 1) * 16 + n] };
            vacc = __builtin_amdgcn_wmma_f32_16x16x4_f32(
                false, wband[t], false, b, (short)0, vacc, false, false);
        }
        res[q] = vacc;
    }

    // ---- elementwise SSIM map + reduction (all res[] share the C/D layout)
    const float C1 = 0.0001f;   // 0.01^2
    const float C2 = 0.0009f;   // 0.03^2
    float lsum = 0.f;
#pragma unroll
    for (int v = 0; v < 8; ++v) {
        float mu1 = res[0][v], mu2 = res[1][v];
        float e11 = res[2][v], e22 = res[3][v], e12 = res[4][v];
        float mu1sq = mu1 * mu1, mu2sq = mu2 * mu2, mu12 = mu1 * mu2;
        float s1  = e11 - mu1sq;
        float s2  = e22 - mu2sq;
        float s12 = e12 - mu12;
        float num = (2.f * mu12 + C1) * (2.f * s12 + C2);
        float den = (mu1sq + mu2sq + C1) * (s1 + s2 + C2);
        lsum += num / den;
    }
#pragma unroll
    for (int off = 16; off > 0; off >>= 1)
        lsum += __shfl_down(lsum, off, 32);
    if (lane == 0)
        atomicAdd(acc, (double)lsum);
}

__global__ void ssim_zero_kernel(double* acc) { *acc = 0.0; }

__global__ void ssim_finalize_kernel(const double* acc, float* out) {
    out[0] = (float)(1.0 - (*acc) * (1.0 / (16.0 * 3.0 * 512.0 * 512.0)));
}

extern "C" void kernel_launch(void* const* d_in, const int* in_sizes, int n_in,
                              void* d_out, int out_size, void* d_ws, size_t ws_size,
                              hipStream_t stream) {
    const float* pred = (const float*)d_in[0];
    const float* targ = (const float*)d_in[1];
    float* out = (float*)d_out;
    double* acc = (double*)d_ws;

    ssim_zero_kernel<<<1, 1, 0, stream>>>(acc);
    ssim_wmma_kernel<<<NBLOCKS, 32 * WAVES_PER_BLOCK, 0, stream>>>(pred, targ, acc);
    ssim_finalize_kernel<<<1, 1, 0, stream>>>(acc, out);
}